// Quanv2d_72584947302548
// MI455X (gfx1250) — compile-verified
//
#include <hip/hip_runtime.h>
#include <math.h>

typedef float v2f __attribute__((ext_vector_type(2)));
typedef float v8f __attribute__((ext_vector_type(8)));

#define NQ   8
#define DIM  256      // 2^NQ state dimension
#define SROW 258      // padded LDS row stride for S (floats)
#define XROW 66       // padded LDS row stride for X staging

// ---------------------------------------------------------------------------
// Kernel 1: build the fixed 256x256 unitary U (product of the 16 controlled
// rotations) column-by-column, storing THREE packed B-operand matrices for
// the 3M complex GEMM:  B_re, B_im, B_sum = B_re + B_im, with
//   Bpk[(k>>1)*256 + n] = float2{ B[k_even][n], B[k_odd][n] },  B[k][n] = U[n][k]
// ---------------------------------------------------------------------------
__global__ __launch_bounds__(64) void build_u_kernel(const float* __restrict__ w,
                                                     float* __restrict__ bre,
                                                     float* __restrict__ bim,
                                                     float* __restrict__ bsm) {
  __shared__ float sre[DIM];
  __shared__ float sim[DIM];
  const int col = blockIdx.x;   // basis state index w  (== k of B)
  const int tid = threadIdx.x;  // 0..63

  for (int j = 0; j < 4; ++j) { sre[tid * 4 + j] = 0.f; sim[tid * 4 + j] = 0.f; }
  __syncthreads();
  if (tid == 0) sre[col] = 1.f;
  __syncthreads();

  for (int g = 0; g < 16; ++g) {
    const int l = g >> 3;          // layer
    const int i = g & 7;           // gate within layer
    const float hh = 0.5f * w[l * 16 + i];
    const float c = cosf(hh);
    const float s = sinf(hh);
    const int pc = 7 - i;                 // control bit (wire i, MSB-first)
    const int pt = 7 - ((i + 1) & 7);     // target bit
    const int lo = pc < pt ? pc : pt;
    const int hi = pc < pt ? pt : pc;
    int x = tid;                          // 6 free bits -> spread around lo,hi
    x = ((x >> lo) << (lo + 1)) | (x & ((1 << lo) - 1));
    x = ((x >> hi) << (hi + 1)) | (x & ((1 << hi) - 1));
    const int i0 = x | (1 << pc);         // control=1, target=0
    const int i1 = i0 | (1 << pt);        // control=1, target=1
    const float a0r = sre[i0], a0i = sim[i0];
    const float a1r = sre[i1], a1i = sim[i1];
    float n0r, n0i, n1r, n1i;
    if (l == 0) {   // RX: [[c, -i s],[-i s, c]]
      n0r = c * a0r + s * a1i;  n0i = c * a0i - s * a1r;
      n1r = c * a1r + s * a0i;  n1i = c * a1i - s * a0r;
    } else {        // RY: [[c, -s],[s, c]]
      n0r = c * a0r - s * a1r;  n0i = c * a0i - s * a1i;
      n1r = s * a0r + c * a1r;  n1i = s * a0i + c * a1i;
    }
    sre[i0] = n0r; sim[i0] = n0i;
    sre[i1] = n1r; sim[i1] = n1i;
    __syncthreads();
  }

  // column w of U -> row k=w of B, packed (even k, odd k) as float2
  const int base = (col >> 1) * DIM;
  const int sel  = col & 1;
  for (int j = 0; j < 4; ++j) {
    const int z = tid * 4 + j;
    bre[(base + z) * 2 + sel] = sre[z];
    bim[(base + z) * 2 + sel] = sim[z];
    bsm[(base + z) * 2 + sel] = sre[z] + sim[z];
  }
}

// ---------------------------------------------------------------------------
// Kernel 2: per block handle 16 batch rows.
//   Phase A: build per-qubit 2-vectors psi from X (H + RZ/RY chain).
//   Phase B: tensor-product expansion -> S[16][256] complex in LDS.
//   Phase C: Y = S * U^T via v_wmma_f32_16x16x4_f32 using the 3M complex
//            decomposition (3 wmma / K-step), then signed |Y|^2 -> <Z_q>.
// ---------------------------------------------------------------------------
__global__ __launch_bounds__(128) void quanv_kernel(const float* __restrict__ X,
                                                    const float* __restrict__ bre,
                                                    const float* __restrict__ bim,
                                                    const float* __restrict__ bsm,
                                                    float* __restrict__ out) {
  __shared__ float smem[32 * SROW + 512];   // Sre | Sim | psi/red
  float* Sre = smem;
  float* Sim = smem + 16 * SROW;
  float* Psi = smem + 32 * SROW;            // 512 floats, reused as reduce buf

  const int tid     = threadIdx.x;          // 0..127
  const int lane    = tid & 31;
  const int wv      = tid >> 5;             // wave 0..3
  const int rowbase = blockIdx.x * 16;

  // ---- stage X tile (16 rows x 64) coalesced into Sre area (padded rows)
  {
    const float* xg = X + rowbase * 64;
    for (int f = tid; f < 1024; f += 128) {
      const int rr = f >> 6, cc = f & 63;
      Sre[rr * XROW + cc] = xg[f];
    }
  }
  __syncthreads();

  // ---- Phase A: one (row, qubit) per thread
  {
    const int r = tid & 15;
    const int q = tid >> 4;                 // 0..7
    float p0r = 0.70710678118f, p0i = 0.f;  // H|0>
    float p1r = 0.70710678118f, p1i = 0.f;
    const int nrot = (q == 7) ? 1 : 9;
    const float* xr = &Sre[r * XROW + q * 9];
    for (int i = 0; i < nrot; ++i) {
      const float hh = 0.5f * xr[i];
      const float c = cosf(hh), s = sinf(hh);
      float n0r, n0i, n1r, n1i;
      if ((i & 1) == 0) {  // RZ: psi0*(c-is), psi1*(c+is)
        n0r = p0r * c + p0i * s;  n0i = p0i * c - p0r * s;
        n1r = p1r * c - p1i * s;  n1i = p1i * c + p1r * s;
      } else {             // RY
        n0r = c * p0r - s * p1r;  n0i = c * p0i - s * p1i;
        n1r = s * p0r + c * p1r;  n1i = s * p0i + c * p1i;
      }
      p0r = n0r; p0i = n0i; p1r = n1r; p1i = n1i;
    }
    float4 v; v.x = p0r; v.y = p0i; v.z = p1r; v.w = p1i;
    *reinterpret_cast<float4*>(&Psi[(q * 16 + r) * 4]) = v;
  }
  __syncthreads();

  // ---- Phase B: expansion. thread -> (row r, 32 amplitudes)
  {
    const int r    = tid & 15;
    const int part = tid >> 4;        // 0..7
    const int h    = part >> 2;       // bit7 of z (wire 0)
    const int jjb  = (part & 3) * 2;  // jj pair handled by this thread

    float pr[8][2][2];
#pragma unroll
    for (int q = 0; q < 8; ++q) {
      const float4 v = *reinterpret_cast<const float4*>(&Psi[(q * 16 + r) * 4]);
      pr[q][0][0] = v.x; pr[q][0][1] = v.y; pr[q][1][0] = v.z; pr[q][1][1] = v.w;
    }
    const float c0r = h ? pr[0][1][0] : pr[0][0][0];
    const float c0i = h ? pr[0][1][1] : pr[0][0][1];

    float hir[8], hii[8];             // wires 0..3 (jj = b6b5b4)
    {
      float t1r[2], t1i[2], t2r[4], t2i[4];
#pragma unroll
      for (int b = 0; b < 2; ++b) {
        t1r[b] = c0r * pr[1][b][0] - c0i * pr[1][b][1];
        t1i[b] = c0r * pr[1][b][1] + c0i * pr[1][b][0];
      }
#pragma unroll
      for (int j = 0; j < 2; ++j)
#pragma unroll
        for (int b = 0; b < 2; ++b) {
          t2r[j * 2 + b] = t1r[j] * pr[2][b][0] - t1i[j] * pr[2][b][1];
          t2i[j * 2 + b] = t1r[j] * pr[2][b][1] + t1i[j] * pr[2][b][0];
        }
#pragma unroll
      for (int j = 0; j < 4; ++j)
#pragma unroll
        for (int b = 0; b < 2; ++b) {
          hir[j * 2 + b] = t2r[j] * pr[3][b][0] - t2i[j] * pr[3][b][1];
          hii[j * 2 + b] = t2r[j] * pr[3][b][1] + t2i[j] * pr[3][b][0];
        }
    }
    float lor[16], loi[16];           // wires 4..7 (kk = b3b2b1b0)
    {
      float t2r[4], t2i[4], t3r[8], t3i[8];
#pragma unroll
      for (int j = 0; j < 2; ++j)
#pragma unroll
        for (int b = 0; b < 2; ++b) {
          t2r[j * 2 + b] = pr[4][j][0] * pr[5][b][0] - pr[4][j][1] * pr[5][b][1];
          t2i[j * 2 + b] = pr[4][j][0] * pr[5][b][1] + pr[4][j][1] * pr[5][b][0];
        }
#pragma unroll
      for (int j = 0; j < 4; ++j)
#pragma unroll
        for (int b = 0; b < 2; ++b) {
          t3r[j * 2 + b] = t2r[j] * pr[6][b][0] - t2i[j] * pr[6][b][1];
          t3i[j * 2 + b] = t2r[j] * pr[6][b][1] + t2i[j] * pr[6][b][0];
        }
#pragma unroll
      for (int j = 0; j < 8; ++j)
#pragma unroll
        for (int b = 0; b < 2; ++b) {
          lor[j * 2 + b] = t3r[j] * pr[7][b][0] - t3i[j] * pr[7][b][1];
          loi[j * 2 + b] = t3r[j] * pr[7][b][1] + t3i[j] * pr[7][b][0];
        }
    }
#pragma unroll
    for (int jj2 = 0; jj2 < 2; ++jj2) {
      const int jj = jjb + jj2;
#pragma unroll
      for (int kk = 0; kk < 16; ++kk) {
        const int z = h * 128 + jj * 16 + kk;
        Sre[r * SROW + z] = hir[jj] * lor[kk] - hii[jj] * loi[kk];
        Sim[r * SROW + z] = hir[jj] * loi[kk] + hii[jj] * lor[kk];
      }
    }
  }
  __syncthreads();

  // ---- Phase C: 3M complex WMMA GEMM + signed-probability reduction.
  {
    const int r    = lane & 15;       // A-fragment row (M)
    const int h    = lane >> 4;       // half-wave -> K pair select
    const int colL = lane & 15;       // B/C fragment column (N low bits)

    const float2* Bre2 = reinterpret_cast<const float2*>(bre);
    const float2* Bim2 = reinterpret_cast<const float2*>(bim);
    const float2* Bsm2 = reinterpret_cast<const float2*>(bsm);

    v8f acc[8];
#pragma unroll
    for (int q = 0; q < 8; ++q) acc[q] = {};
    float sg[4];
#pragma unroll
    for (int q = 4; q < 8; ++q)       // z bits 3..0 live in colL
      sg[q - 4] = ((colL >> (7 - q)) & 1) ? -1.f : 1.f;

    for (int tt = 0; tt < 4; ++tt) {          // this wave's 4 column tiles
      const int t  = wv * 4 + tt;
      const int n0 = t * 16;
      v8f p1 = {}, p2 = {}, p3 = {};          // 3 independent wmma chains
      const float*  aR = &Sre[r * SROW + 2 * h];
      const float*  aI = &Sim[r * SROW + 2 * h];
      const float2* bR = &Bre2[h * DIM + n0 + colL];
      const float2* bI = &Bim2[h * DIM + n0 + colL];
      const float2* bS = &Bsm2[h * DIM + n0 + colL];
#pragma unroll 4
      for (int ks = 0; ks < 64; ++ks) {       // K = 256 in steps of 4
        const float2 arf = *reinterpret_cast<const float2*>(aR + ks * 4);
        const float2 aif = *reinterpret_cast<const float2*>(aI + ks * 4);
        const float2 brf = bR[ks * 2 * DIM];
        const float2 bif = bI[ks * 2 * DIM];
        const float2 bsf = bS[ks * 2 * DIM];
        v2f ar2 = {arf.x, arf.y};
        v2f ai2 = {aif.x, aif.y};
        v2f as2 = {arf.x + aif.x, arf.y + aif.y};
        v2f br2 = {brf.x, brf.y};
        v2f bi2 = {bif.x, bif.y};
        v2f bs2 = {bsf.x, bsf.y};
        p1 = __builtin_amdgcn_wmma_f32_16x16x4_f32(false, ar2, false, br2,
                                                   (short)0, p1, false, false);
        p2 = __builtin_amdgcn_wmma_f32_16x16x4_f32(false, ai2, false, bi2,
                                                   (short)0, p2, false, false);
        p3 = __builtin_amdgcn_wmma_f32_16x16x4_f32(false, as2, false, bs2,
                                                   (short)0, p3, false, false);
      }
      const v8f yre = p1 - p2;                // 3M recombination
      const v8f yim = p3 - p1 - p2;
      const v8f p = yre * yre + yim * yim;    // probabilities for this tile
#pragma unroll
      for (int q = 0; q < 4; ++q) {           // z bits 7..4 live in t
        const float s = ((t >> (3 - q)) & 1) ? -1.f : 1.f;
        acc[q] += s * p;
      }
#pragma unroll
      for (int q = 4; q < 8; ++q) acc[q] += sg[q - 4] * p;
    }

    // reduce over the 16 columns held by each half-wave
#pragma unroll
    for (int off = 1; off <= 8; off <<= 1)
#pragma unroll
      for (int q = 0; q < 8; ++q)
#pragma unroll
        for (int v = 0; v < 8; ++v)
          acc[q][v] += __shfl_xor(acc[q][v], off, 32);

    // per-wave partials -> LDS (reuse Psi region: 4 waves * 16 rows * 8 q)
#pragma unroll
    for (int q = 0; q < 8; ++q) {
      if (colL == q) {
#pragma unroll
        for (int v = 0; v < 8; ++v)
          Psi[wv * 128 + (h * 8 + v) * 8 + q] = acc[q][v];
      }
    }
  }
  __syncthreads();

  // ---- combine the 4 waves' column partitions, write out (16 rows x 8)
  {
    const int row = tid >> 3;
    const int q   = tid & 7;
    float v = 0.f;
#pragma unroll
    for (int w = 0; w < 4; ++w) v += Psi[w * 128 + row * 8 + q];
    out[(rowbase + row) * 8 + q] = v;
  }
}

// ---------------------------------------------------------------------------
extern "C" void kernel_launch(void* const* d_in, const int* in_sizes, int n_in,
                              void* d_out, int out_size, void* d_ws, size_t ws_size,
                              hipStream_t stream) {
  const float* X = (const float*)d_in[0];
  const float* w = (const float*)d_in[1];
  float* bre = (float*)d_ws;          // 65536 floats (256 KB), packed B_re
  float* bim = bre + DIM * DIM;       // 65536 floats (256 KB), packed B_im
  float* bsm = bim + DIM * DIM;       // 65536 floats (256 KB), packed B_re+B_im

  build_u_kernel<<<DIM, 64, 0, stream>>>(w, bre, bim, bsm);

  const int nrows  = in_sizes[0] / 64;   // 16384
  const int blocks = nrows / 16;         // 1024
  quanv_kernel<<<blocks, 128, 0, stream>>>(X, bre, bim, bsm, (float*)d_out);
}